// BlockExtractor_34522947125556
// MI455X (gfx1250) — compile-verified
//
#include <hip/hip_runtime.h>
#include <math.h>

// Problem shape (from setup_inputs): source (8,64,128,128) f32, flow (8,2,128,128) f32, K=3.
#define B_  8
#define C_  64
#define H_  128
#define W_  128

// Native clang vectors (HIP_vector_type classes are rejected by the NT builtins).
typedef float v4f __attribute__((ext_vector_type(4)));
// Horizontal corner pair: only 4-byte aligned (xL arbitrary), so declare align(4).
typedef float v2f __attribute__((ext_vector_type(2), aligned(4)));

__global__ __launch_bounds__(256) void block_extract_kernel(
    const float* __restrict__ src,    // [B, C, H, W]
    const float* __restrict__ flow,   // [B, 2, H, W]
    float* __restrict__ out,          // [B, C, K*H, K*W]
    int K, int KH, int KW, int nquads)
{
    const int q = blockIdx.x * blockDim.x + threadIdx.x;
    if (q >= nquads) return;

    const int qpr = KW >> 2;                 // float4 quads per output row
    const int x0  = (q % qpr) << 2;
    const int y   = (q / qpr) % KH;
    const int b   =  q / (qpr * KH);
    // Quads-per-batch = 4096*K^2 is always a multiple of blockDim (256), so a
    // workgroup never straddles batches: b is wave/block-uniform. Making it a
    // scalar enables SADDR-based (SGPR base + VGPR offset) global loads.
    const int bu  = __builtin_amdgcn_readfirstlane(b);

    const int rad = (K - 1) >> 1;
    const int yf  = y / K;
    const int yo  = y - yf * K - rad;        // (y % K) - rad

    const int HW = H_ * W_;
    const float* __restrict__ flowb = flow + (size_t)bu * 2 * HW;

    // Per-pixel state, computed once and reused across all 64 channels:
    //   oT/oB : element offset of the (possibly left-shifted) horizontal pair
    //   wT/wB : packed {left,right} bilinear weights with the right-edge clamp
    //           folded in, so the inner loop is pk_mul + pk_fma + h-add.
    int oT[4], oB[4];
    v2f wT[4], wB[4];

#pragma unroll
    for (int j = 0; j < 4; ++j) {
        const int x  = x0 + j;
        const int xf = x / K;
        const int xo = x - xf * K - rad;

        const float fy = flowb[yf * W_ + xf];          // flow channel 0 (y)
        const float fx = flowb[HW + yf * W_ + xf];     // flow channel 1 (x)

        const float ys = (float)(yf + yo) + fy;
        const float xs = (float)(xf + xo) + fx;

        int yT = (int)floorf(ys);
        yT = min(max(yT, 0), H_ - 1);
        const int yB = min(yT + 1, H_ - 1);
        int xL = (int)floorf(xs);
        xL = min(max(xL, 0), W_ - 1);

        const float dy = ys - (float)yT;               // delta vs clipped floor
        const float dx = xs - (float)xL;               // (reference semantics)

        const float wTL = (1.0f - dy) * (1.0f - dx);
        const float wTR = (1.0f - dy) * dx;
        const float wBL = dy * (1.0f - dx);
        const float wBR = dy * dx;

        // Right-edge clamp: xR==xL==W-1. Shift the 2-wide load window left one
        // column and move all horizontal weight onto the .y lane of the pair.
        const bool hi  = (xL == W_ - 1);
        const int  xLs = xL - (hi ? 1 : 0);
        oT[j] = yT * W_ + xLs;
        oB[j] = yB * W_ + xLs;
        wT[j].x = hi ? 0.0f : wTL;
        wT[j].y = hi ? (wTL + wTR) : wTR;
        wB[j].x = hi ? 0.0f : wBL;
        wB[j].y = hi ? (wBL + wBR) : wBR;
    }

    const float* __restrict__ plane0 = src + (size_t)bu * C_ * HW;   // scalar base
    float* __restrict__ op0 = out + ((size_t)bu * C_ * KH + y) * (size_t)KW + x0;
    const size_t ostride = (size_t)KH * KW;

    // Fully unrolled channel loop: c*HW*4 = c*64KB <= 4MB folds into the
    // signed 24-bit instruction offset of each global load -> no per-channel
    // address arithmetic in the steady state.
#pragma unroll
    for (int c = 0; c < C_; ++c) {
        const float* __restrict__ pl = plane0 + c * HW;

        // Lookahead prefetch of a later channel plane (gfx1250 global_prefetch_b8);
        // condition is constant-folded by the full unroll.
        if ((c & 15) == 0 && c + 16 < C_) {
            __builtin_prefetch(pl + 16 * HW + oT[0], 0, 1);
            __builtin_prefetch(pl + 16 * HW + oB[3], 0, 1);
        }

        float r[4];
#pragma unroll
        for (int j = 0; j < 4; ++j) {
            const v2f vT = *(const v2f*)(pl + oT[j]);   // {tl,tr} in one b64 gather
            const v2f vB = *(const v2f*)(pl + oB[j]);   // {bl,br} in one b64 gather
            // v_pk_mul_f32 + v_pk_fma_f32 on the 64-bit pairs, then one h-add.
            const v2f acc = __builtin_elementwise_fma(vT, wT[j], vB * wB[j]);
            r[j] = acc.x + acc.y;
        }

        // Write-once 302MB output: non-temporal 128-bit coalesced store so it
        // does not evict the L2-resident 32MB source working set.
        v4f v = { r[0], r[1], r[2], r[3] };
        __builtin_nontemporal_store(v, (v4f*)(op0 + (size_t)c * ostride));
    }
}

extern "C" void kernel_launch(void* const* d_in, const int* in_sizes, int n_in,
                              void* d_out, int out_size, void* d_ws, size_t ws_size,
                              hipStream_t stream) {
    const float* src  = (const float*)d_in[0];
    const float* flow = (const float*)d_in[1];
    float* out = (float*)d_out;

    // out_size = B*C*(K*H)*(K*W) = (B*C*H*W) * K^2  ->  recover K.
    int K = 3;
    const long long base = (long long)B_ * C_ * H_ * W_;
    if (out_size > 0 && base > 0) {
        const long long ratio = (long long)out_size / base;
        for (int k = 1; k <= 16; ++k)
            if ((long long)k * k == ratio) { K = k; break; }
    }
    const int KH = K * H_;
    const int KW = K * W_;
    const int nquads = B_ * KH * (KW / 4);   // 4 output pixels (one float4) per thread

    const int block = 256;                    // 8 wave32s per workgroup
    const int grid  = (nquads + block - 1) / block;
    block_extract_kernel<<<grid, block, 0, stream>>>(src, flow, out, K, KH, KW, nquads);
}